// G2AEmbedNet_78941498901021
// MI455X (gfx1250) — compile-verified
//
#include <hip/hip_runtime.h>
#include <math.h>

// ---------------------------------------------------------------------------
// G2AEmbedNet for MI455X (gfx1250, wave32, WMMA).
// Memory-bound on the 201MB obs read -> stream obs once through f32 WMMA.
// GRU is dead code except sequences s=0,1 of batch 0 (hw = y[:3]).
// ---------------------------------------------------------------------------

#define N_THREAD 16384
#define N_AGENT  3
#define OBS_DIM  1024
#define HIDDEN   64
#define ATTN     32
#define TAU      0.01f
#define ROWS     (N_THREAD * N_AGENT)   // 49152

typedef float v2f __attribute__((ext_vector_type(2)));
typedef float v8f __attribute__((ext_vector_type(8)));

// ===========================================================================
// Kernel 1: h = relu(obs @ enc_w^T + enc_b)  [ROWS x 64], fused with
//           q = h@q_w^T, k = h@k_w^T, v = relu(h@v_w^T + v_b)  [ROWS x 32 each]
// One wave computes a 16-row x 64-col tile with V_WMMA_F32_16X16X4_F32.
// enc_w chunks staged in LDS; the finished h tile is parked in per-wave LDS
// and re-read as the A operand for the qkv WMMAs (no global round trip).
// ===========================================================================
__global__ __launch_bounds__(128) void enc_qkv_kernel(
    const float* __restrict__ obs,  const float* __restrict__ enc_w,
    const float* __restrict__ enc_b,
    const float* __restrict__ q_w,  const float* __restrict__ k_w,
    const float* __restrict__ v_w,  const float* __restrict__ v_b,
    float* __restrict__ h_out, float* __restrict__ q_out,
    float* __restrict__ k_out, float* __restrict__ v_out)
{
    __shared__ float Wlds[64][128];     // enc_w[:, kb:kb+128]  (32 KB)
    __shared__ float hlds[4][16][66];   // per-wave h tile, padded vs bank conflicts

    const int tid  = threadIdx.x;
    const int lane = tid & 31;
    const int wave = tid >> 5;
    const int nn   = lane & 15;         // M row (A) / N col (B,C)
    const int hi   = lane >> 4;         // half-wave select
    const int koff = hi * 2;            // K offset per ISA f32 A/B striping
    const int row0 = (blockIdx.x * 4 + wave) * 16;

    v8f acc[4] = {};                    // 16x64 C tile (4 x 16x16)

    for (int kb = 0; kb < OBS_DIM; kb += 128) {
        __syncthreads();
        // stage enc_w chunk (coalesced float4)
        for (int idx = tid; idx < 64 * 32; idx += 128) {
            int n = idx >> 5, kq = (idx & 31) * 4;
            *(float4*)&Wlds[n][kq] = *(const float4*)&enc_w[(size_t)n * OBS_DIM + kb + kq];
        }
        __syncthreads();
        if (kb + 128 < OBS_DIM)
            __builtin_prefetch(&obs[(size_t)(row0 + nn) * OBS_DIM + kb + 128 + koff], 0, 0);

        for (int ks = 0; ks < 128; ks += 4) {
            // A frag: lanes 0-15 rows M, K = k0/k0+1 ; lanes 16-31 K = k0+2/k0+3
            v2f a = *(const v2f*)&obs[(size_t)(row0 + nn) * OBS_DIM + kb + ks + koff];
            #pragma unroll
            for (int nt = 0; nt < 4; ++nt) {
                // B frag (4x16, K x N): B[k][n] = enc_w[nt*16+n][k]
                v2f b = *(const v2f*)&Wlds[nt * 16 + nn][ks + koff];
                acc[nt] = __builtin_amdgcn_wmma_f32_16x16x4_f32(
                    false, a, false, b, (short)0, acc[nt], false, false);
            }
        }
    }

    // epilogue: bias + relu, store h to global and to per-wave LDS
    #pragma unroll
    for (int nt = 0; nt < 4; ++nt) {
        float bias = enc_b[nt * 16 + nn];
        #pragma unroll
        for (int g = 0; g < 8; ++g) {
            float vv = acc[nt][g] + bias;
            vv = vv > 0.f ? vv : 0.f;
            int mrow = g + 8 * hi;                  // C layout: VGPR g -> M=g / M=g+8
            int col  = nt * 16 + nn;
            h_out[(size_t)(row0 + mrow) * HIDDEN + col] = vv;
            hlds[wave][mrow][col] = vv;
        }
    }
    __syncthreads();

    // fused qkv: C[16x96] = h_tile(16x64) x [q_w|k_w|v_w]^T
    v8f aq[6] = {};
    for (int ks = 0; ks < HIDDEN; ks += 4) {
        v2f a = *(const v2f*)&hlds[wave][nn][ks + koff];
        #pragma unroll
        for (int t = 0; t < 6; ++t) {
            const float* wsrc = (t < 2) ? q_w : (t < 4) ? k_w : v_w;
            int rowoff = (t & 1) * 16;
            v2f b = *(const v2f*)&wsrc[(size_t)(rowoff + nn) * HIDDEN + ks + koff];
            aq[t] = __builtin_amdgcn_wmma_f32_16x16x4_f32(
                false, a, false, b, (short)0, aq[t], false, false);
        }
    }
    #pragma unroll
    for (int t = 0; t < 6; ++t) {
        int col = (t & 1) * 16 + nn;
        float bias = (t >= 4) ? v_b[col] : 0.f;
        float* dst = (t < 2) ? q_out : (t < 4) ? k_out : v_out;
        #pragma unroll
        for (int g = 0; g < 8; ++g) {
            float vv = aq[t][g] + bias;
            if (t >= 4) vv = vv > 0.f ? vv : 0.f;
            dst[(size_t)(row0 + g + 8 * hi) * ATTN + col] = vv;
        }
    }
}

// ===========================================================================
// Kernel 2: bi-GRU + hard gating, ONLY for the live sequences (s=0, s=1, b=0).
// hw = softmax((logits[:3] + gumbel)/TAU). One block, wave s handles seq s;
// lane l owns hidden units {l, l+32}.
// ===========================================================================
__device__ __forceinline__ float hash_u01(unsigned x) {
    x = x * 0x9E3779B9u + 0x85EBCA6Bu;
    x ^= x >> 16; x *= 0x7feb352du; x ^= x >> 15; x *= 0x846ca68bu; x ^= x >> 16;
    float u = (float)(x >> 8) * (1.0f / 16777216.0f);
    return fminf(fmaxf(u, 1e-10f), 1.0f - 1e-7f);
}

__global__ __launch_bounds__(64) void gru_gate_kernel(
    const float* __restrict__ h,
    const float* __restrict__ wihf, const float* __restrict__ whhf,
    const float* __restrict__ bihf, const float* __restrict__ bhhf,
    const float* __restrict__ wihb, const float* __restrict__ whhb,
    const float* __restrict__ bihb, const float* __restrict__ bhhb,
    const float* __restrict__ hard_w, const float* __restrict__ hard_b,
    float* __restrict__ hw_out)
{
    __shared__ float pairs[2][2][128];   // [seq][t][concat(h_i,h_j)]
    __shared__ float hst[2][64];
    __shared__ float outf[2][2][64];
    __shared__ float outb[2][2][64];

    const int tid = threadIdx.x, lane = tid & 31, s = tid >> 5;
    const int nbt[2][2] = {{1, 2}, {0, 2}};   // neighbors of agents 0,1 (b=0)

    for (int idx = tid; idx < 512; idx += 64) {
        int ss = idx >> 8, t = (idx >> 7) & 1, c = idx & 127;
        pairs[ss][t][c] = (c < 64) ? h[ss * 64 + c]
                                   : h[nbt[ss][t] * 64 + (c - 64)];
    }
    __syncthreads();

    for (int dir = 0; dir < 2; ++dir) {
        const float* Wih = dir ? wihb : wihf;
        const float* Whh = dir ? whhb : whhf;
        const float* bih = dir ? bihb : bihf;
        const float* bhh = dir ? bhhb : bhhf;
        hst[s][lane] = 0.f; hst[s][lane + 32] = 0.f;
        __syncthreads();

        for (int step = 0; step < 2; ++step) {
            int tpos = dir ? (1 - step) : step;       // backward scans reversed input
            const float* xin = pairs[s][tpos];
            float hnew[2];
            #pragma unroll
            for (int w2 = 0; w2 < 2; ++w2) {
                int u = lane + 32 * w2;
                float ir = bih[u], iz = bih[64 + u], inn = bih[128 + u];
                for (int c = 0; c < 128; ++c) {
                    float x = xin[c];
                    ir  += Wih[(size_t)u * 128 + c] * x;
                    iz  += Wih[(size_t)(64 + u) * 128 + c] * x;
                    inn += Wih[(size_t)(128 + u) * 128 + c] * x;
                }
                float hr = bhh[u], hz = bhh[64 + u], hn = bhh[128 + u];
                for (int c = 0; c < 64; ++c) {
                    float hh = hst[s][c];
                    hr += Whh[(size_t)u * 64 + c] * hh;
                    hz += Whh[(size_t)(64 + u) * 64 + c] * hh;
                    hn += Whh[(size_t)(128 + u) * 64 + c] * hh;
                }
                float r  = 1.f / (1.f + __expf(-(ir + hr)));
                float z  = 1.f / (1.f + __expf(-(iz + hz)));
                float nv = tanhf(inn + r * hn);
                hnew[w2] = (1.f - z) * nv + z * hst[s][u];
            }
            __syncthreads();
            hst[s][lane] = hnew[0]; hst[s][lane + 32] = hnew[1];
            float* oarr = dir ? &outb[s][tpos][0] : &outf[s][tpos][0];
            oarr[lane] = hnew[0]; oarr[lane + 32] = hnew[1];
            __syncthreads();
        }
    }

    if (tid < 3) {            // flat rows 0..2 = (s,t) = (0,0),(0,1),(1,0)
        int row = tid, ss = row >> 1, t = row & 1;
        float lg[2];
        for (int c = 0; c < 2; ++c) {
            float a = hard_b[c];
            for (int d = 0; d < 64; ++d) a += hard_w[c * 128 + d] * outf[ss][t][d];
            for (int d = 0; d < 64; ++d) a += hard_w[c * 128 + 64 + d] * outb[ss][t][d];
            float u = hash_u01((unsigned)(row * 2 + c));   // stand-in for jax key(42)
            float g = -logf(-logf(u));
            lg[c] = (a + g) / TAU;
        }
        float m = fmaxf(lg[0], lg[1]);
        float e0 = __expf(lg[0] - m), e1 = __expf(lg[1] - m);
        float inv = 1.f / (e0 + e1);
        hw_out[row * 2 + 0] = e0 * inv;
        hw_out[row * 2 + 1] = e1 * inv;
    }
}

// ===========================================================================
// Kernel 3: attention + output assembly. One thread per (b, i).
// out[b,i,0:64]=h ; x part applies the reference's (0,2,1) transpose scatter.
// ===========================================================================
__global__ __launch_bounds__(256) void attn_out_kernel(
    const float* __restrict__ h, const float* __restrict__ q,
    const float* __restrict__ k, const float* __restrict__ v,
    const float* __restrict__ hw, float* __restrict__ out)
{
    int g = blockIdx.x * blockDim.x + threadIdx.x;
    if (g >= ROWS) return;
    int b = g / 3, i = g % 3;
    const int nb0[3] = {1, 0, 0}, nb1[3] = {2, 2, 1};
    int j1 = nb0[i], j2 = nb1[i];

    const float* qi = q + (size_t)g * ATTN;
    const float* k1 = k + (size_t)(b * 3 + j1) * ATTN;
    const float* k2 = k + (size_t)(b * 3 + j2) * ATTN;
    float s1 = 0.f, s2 = 0.f;
    for (int a = 0; a < ATTN; a += 4) {
        float4 qa = *(const float4*)&qi[a];
        float4 ka = *(const float4*)&k1[a];
        float4 kb = *(const float4*)&k2[a];
        s1 += qa.x * ka.x + qa.y * ka.y + qa.z * ka.z + qa.w * ka.w;
        s2 += qa.x * kb.x + qa.y * kb.y + qa.z * kb.z + qa.w * kb.w;
    }
    const float isq = 0.17677669529663687f;     // 1/sqrt(32)
    s1 *= isq; s2 *= isq;
    float m = fmaxf(s1, s2);
    float e1 = __expf(s1 - m), e2 = __expf(s2 - m);
    float inv = 1.f / (e1 + e2);
    float w1 = e1 * inv * hw[i * 2 + 0];
    float w2 = e2 * inv * hw[i * 2 + 1];

    const float* v1 = v + (size_t)(b * 3 + j1) * ATTN;
    const float* v2 = v + (size_t)(b * 3 + j2) * ATTN;
    float* ob = out + (size_t)b * 3 * 96;
    for (int a = 0; a < ATTN; ++a) {
        float xv = v1[a] * w1 + v2[a] * w2;
        int p = 3 * a + i;                       // inverse of (0,2,1) reshape
        ob[(p >> 5) * 96 + 64 + (p & 31)] = xv;
    }
    const float* hr = h + (size_t)g * HIDDEN;
    float* oh = out + (size_t)g * 96;
    for (int c = 0; c < HIDDEN; c += 4)
        *(float4*)&oh[c] = *(const float4*)&hr[c];
}

// ===========================================================================
extern "C" void kernel_launch(void* const* d_in, const int* in_sizes, int n_in,
                              void* d_out, int out_size, void* d_ws, size_t ws_size,
                              hipStream_t stream)
{
    const float* obs    = (const float*)d_in[0];
    const float* enc_w  = (const float*)d_in[1];
    const float* enc_b  = (const float*)d_in[2];
    const float* wihf   = (const float*)d_in[3];
    const float* whhf   = (const float*)d_in[4];
    const float* bihf   = (const float*)d_in[5];
    const float* bhhf   = (const float*)d_in[6];
    const float* wihb   = (const float*)d_in[7];
    const float* whhb   = (const float*)d_in[8];
    const float* bihb   = (const float*)d_in[9];
    const float* bhhb   = (const float*)d_in[10];
    const float* hard_w = (const float*)d_in[11];
    const float* hard_b = (const float*)d_in[12];
    const float* q_w    = (const float*)d_in[13];
    const float* k_w    = (const float*)d_in[14];
    const float* v_w    = (const float*)d_in[15];
    const float* v_b    = (const float*)d_in[16];

    float* ws = (float*)d_ws;                    // ~31.5 MB used
    float* h  = ws;
    float* qb = h  + (size_t)ROWS * HIDDEN;
    float* kb = qb + (size_t)ROWS * ATTN;
    float* vb = kb + (size_t)ROWS * ATTN;
    float* hw = vb + (size_t)ROWS * ATTN;

    enc_qkv_kernel<<<ROWS / 64, 128, 0, stream>>>(
        obs, enc_w, enc_b, q_w, k_w, v_w, v_b, h, qb, kb, vb);
    gru_gate_kernel<<<1, 64, 0, stream>>>(
        h, wihf, whhf, bihf, bhhf, wihb, whhb, bihb, bhhb, hard_w, hard_b, hw);
    attn_out_kernel<<<(ROWS + 255) / 256, 256, 0, stream>>>(
        h, qb, kb, vb, hw, (float*)d_out);
}